// ParallelShortHyenaOperator_481036337263
// MI455X (gfx1250) — compile-verified
//
#include <hip/hip_runtime.h>
#include <cstdint>

// Hyena short-conv operator, MI455X (gfx1250).
// Pure HBM-streaming kernel: ~1.07 GB traffic -> ~46us floor @ 23.3 TB/s.
// Each thread: 4 contiguous channels (float4) x 128 timesteps; K=3 causal
// taps carried in registers. Coefficients staged to LDS via CDNA5 async
// global->LDS copies; main streams use non-temporal b128 loads/stores plus
// global_prefetch_b8 lookahead.

typedef float f4 __attribute__((ext_vector_type(4)));

#define B_c 2
#define L_c 8192
#define G_c 256
#define DG_c 16
#define K_c 3
#define D_c (G_c * DG_c)          // 4096
#define TT_c 128                  // timesteps per thread chunk
#define TPB_c 256                 // threads per block (8 wave32s)
#define QGROUPS_c ((D_c / 4) / TPB_c)  // 4 quad-groups over D
#define NCHUNK_c (L_c / TT_c)          // 64 chunks over L
#define PF_c 8                         // prefetch distance (timesteps)
#define SM_BIAS_c (TPB_c * 4)          // 1024 floats of bias slice
#define SM_H_c ((TPB_c / 4) * K_c)     // 192 floats of filter slice

static_assert(QGROUPS_c * TPB_c * 4 == D_c, "quad coverage");
static_assert(NCHUNK_c * TT_c == L_c, "chunk coverage");

__global__ __launch_bounds__(TPB_c)
void hyena_short_conv_kernel(const float* __restrict__ x1,
                             const float* __restrict__ x2,
                             const float* __restrict__ v,
                             const float* __restrict__ h,
                             const float* __restrict__ bias,
                             float* __restrict__ out)
{
    __shared__ float smem[SM_BIAS_c + SM_H_c];

    const int tid = threadIdx.x;
    const int blk = blockIdx.x;
    const int qg  = blk % QGROUPS_c;                 // which D quad-group
    const int c   = (blk / QGROUPS_c) % NCHUNK_c;    // which L chunk
    const int b   = blk / (QGROUPS_c * NCHUNK_c);    // batch

    // ---- Stage this block's bias slice (4KB) + filter slice (768B) into LDS
    //      with CDNA5 async global->LDS b128 copies (tracked by ASYNCcnt). ----
    const uint32_t sbase = (uint32_t)(uintptr_t)&smem[0]; // LDS byte offset
    {
        const uint64_t g = (uint64_t)(uintptr_t)(bias + qg * SM_BIAS_c + tid * 4);
        const uint32_t l = sbase + tid * 16;
        asm volatile("global_load_async_to_lds_b128 %0, %1, off"
                     :: "v"(l), "v"(g) : "memory");
    }
    if (tid < SM_H_c / 4) {
        const uint64_t g = (uint64_t)(uintptr_t)(h + qg * SM_H_c + tid * 4);
        const uint32_t l = sbase + SM_BIAS_c * 4 + tid * 16;
        asm volatile("global_load_async_to_lds_b128 %0, %1, off"
                     :: "v"(l), "v"(g) : "memory");
    }
    asm volatile("s_wait_asynccnt 0" ::: "memory");
    __syncthreads();

    // Per-thread coefficients from LDS (ds_load). 4 consecutive channels of a
    // 4-aligned quad always lie in one group of 16, so one (w0,w1,w2) each.
    const f4 bias4 = *reinterpret_cast<const f4*>(&smem[tid * 4]);
    const int gl = tid >> 2;                         // group local to block
    const float w0 = smem[SM_BIAS_c + gl * 3 + 0];
    const float w1 = smem[SM_BIAS_c + gl * 3 + 1];
    const float w2 = smem[SM_BIAS_c + gl * 3 + 2];

    const int  d0   = (qg * TPB_c + tid) * 4;        // first channel of quad
    const int  t0   = c * TT_c;                      // first timestep of chunk
    const long base = (long)b * L_c * D_c + d0;

    // Causal-pad halo: z at t0-1, t0-2 (zero for the first chunk).
    f4 zm1 = {0.f, 0.f, 0.f, 0.f};
    f4 zm2 = {0.f, 0.f, 0.f, 0.f};
    if (t0 > 0) {
        const long i1 = base + (long)(t0 - 1) * D_c;
        const long i2 = base + (long)(t0 - 2) * D_c;
        zm1 = __builtin_nontemporal_load((const f4*)(x2 + i1)) *
              __builtin_nontemporal_load((const f4*)(v  + i1));
        zm2 = __builtin_nontemporal_load((const f4*)(x2 + i2)) *
              __builtin_nontemporal_load((const f4*)(v  + i2));
    }

    const float* p1 = x1 + base + (long)t0 * D_c;
    const float* p2 = x2 + base + (long)t0 * D_c;
    const float* pv = v  + base + (long)t0 * D_c;
    float*       po = out + base + (long)t0 * D_c;

#pragma unroll 4
    for (int i = 0; i < TT_c; ++i) {
        // Speculative lookahead (global_prefetch_b8); OOB at the tail is
        // silently dropped by hardware (speculative TH).
        __builtin_prefetch(p1 + (long)PF_c * D_c, 0, 0);
        __builtin_prefetch(p2 + (long)PF_c * D_c, 0, 0);
        __builtin_prefetch(pv + (long)PF_c * D_c, 0, 0);

        const f4 a = __builtin_nontemporal_load((const f4*)p1);
        const f4 z = __builtin_nontemporal_load((const f4*)p2) *
                     __builtin_nontemporal_load((const f4*)pv);

        f4 y = w2 * z + bias4;
        y += w1 * zm1;
        y += w0 * zm2;
        __builtin_nontemporal_store(a * y, (f4*)po);

        zm2 = zm1;
        zm1 = z;
        p1 += D_c; p2 += D_c; pv += D_c; po += D_c;
    }
}

extern "C" void kernel_launch(void* const* d_in, const int* in_sizes, int n_in,
                              void* d_out, int out_size, void* d_ws, size_t ws_size,
                              hipStream_t stream) {
    (void)in_sizes; (void)n_in; (void)d_ws; (void)ws_size; (void)out_size;
    const float* x1   = (const float*)d_in[0];
    const float* x2   = (const float*)d_in[1];
    const float* v    = (const float*)d_in[2];
    const float* h    = (const float*)d_in[3];
    const float* bias = (const float*)d_in[4];
    float* out = (float*)d_out;

    dim3 grid(QGROUPS_c * NCHUNK_c * B_c);  // 512 blocks
    dim3 block(TPB_c);                      // 256 threads = 8 wave32s
    hipLaunchKernelGGL(hyena_short_conv_kernel, grid, block, 0, stream,
                       x1, x2, v, h, bias, out);
}